// VectorQuantizer_24017457119610
// MI455X (gfx1250) — compile-verified
//
#include <hip/hip_runtime.h>

typedef __attribute__((ext_vector_type(2))) float v2f;
typedef __attribute__((ext_vector_type(8))) float v8f;

#define KMAX 1024            // max codebook entries staged in LDS (K == 1024 here)
#define WAVES_PER_BLOCK 8
#define TILES_PER_WAVE 8

__global__ __launch_bounds__(256) void vq_wmma_kernel(
    const float* __restrict__ x,         // [N,2]
    const float* __restrict__ codebook,  // [K,2]
    float* __restrict__ out_e,           // [N,2]
    float* __restrict__ out_ids,         // [N] (ids as float)
    int N, int K)
{
    // B operand pre-swizzled into the exact per-lane WMMA layout.
    // For k-tile t (16 codes), lane L, the lane's 2 floats (B rows K0..K3):
    //   L <  16 : ( -2*c0[k],  -2*c1[k] )      k = t*16 + L
    //   L >= 16 : ( ||c[k]||^2,  1.0f    )      k = t*16 + (L-16)
    // ldsB[t*64 + L*2 + {0,1}]  -> one ds_load_b64 per lane per tile, no fixup.
    __shared__ float ldsB[(KMAX / 16) * 64];

    const int tid = threadIdx.x;
    for (int k = tid; k < K; k += blockDim.x) {
        float c0 = codebook[2 * k + 0];
        float c1 = codebook[2 * k + 1];
        int t = k >> 4, l = k & 15;
        float* base = ldsB + t * 64;
        base[2 * l + 0]      = -2.0f * c0;
        base[2 * l + 1]      = -2.0f * c1;
        base[32 + 2 * l + 0] = c0 * c0 + c1 * c1;
        base[32 + 2 * l + 1] = 1.0f;
    }
    __syncthreads();

    const int lane = tid & 31;
    const int wave = tid >> 5;
    const int col  = lane & 15;     // D-matrix column owned by this lane
    const bool hi  = lane >= 16;    // lanes 16..31: A supplies K2/K3, D rows 8..15

    const v2f* bvec = (const v2f*)ldsB + lane;   // lane's B slot, stride 32 v2f/tile
    const int  KT   = K >> 4;
    const int  tileBase = (blockIdx.x * WAVES_PER_BLOCK + wave) * TILES_PER_WAVE;

    for (int j = 0; j < TILES_PER_WAVE; ++j) {
        const int tile = tileBase + j;           // uniform across wave
        const int row0 = tile * 16;
        if (row0 >= N) break;                    // uniform branch: EXEC stays full

        // A row m = (x0, x1, 1, ||x||^2); with B col n = (-2c0, -2c1, ||c||^2, 1)
        // the WMMA yields the full distance  ||x||^2 - 2 x.c + ||c||^2.
        int r = row0 + col;
        if (r >= N) r = N - 1;                   // safety clamp (N % 16 == 0 here)
        v2f xv = *(const v2f*)(x + 2 * r);
        float nrm = xv.x * xv.x + xv.y * xv.y;
        v2f a;
        a.x = hi ? 1.0f : xv.x;
        a.y = hi ? nrm  : xv.y;

        float minv[8];
        int   mini[8];
#pragma unroll
        for (int q = 0; q < 8; ++q) { minv[q] = 3.4e38f; mini[q] = 0; }

#pragma unroll 2
        for (int t = 0; t < KT; ++t) {
            v2f b = bvec[t * 32];                // ds_load_b64

            v8f c = {};
            v8f d = __builtin_amdgcn_wmma_f32_16x16x4_f32(
                false, a, false, b, (short)0, c, false, false);

            const int kk = t * 16 + col;         // candidate code index for this lane
#pragma unroll
            for (int q = 0; q < 8; ++q) {
                float dv = d[q];
                if (dv < minv[q]) { minv[q] = dv; mini[q] = kk; }
            }
        }

        // Branchless butterfly min-reduce across the 16 lanes of each half
        // (masks <= 8 keep lanes 0-15 and 16-31 independent). Tie-break:
        // smaller index (matches jnp.argmin first-occurrence semantics).
#pragma unroll
        for (int q = 0; q < 8; ++q) {
            float v = minv[q];
            int   i = mini[q];
#pragma unroll
            for (int m = 8; m >= 1; m >>= 1) {
                float ov = __shfl_xor(v, m, 32);
                int   oi = __shfl_xor(i, m, 32);
                int better = (int)(ov < v) | ((int)(ov == v) & (int)(oi < i));
                v = better ? ov : v;
                i = better ? oi : i;
            }
            minv[q] = v;
            mini[q] = i;
        }

        // Lane 0 writes rows row0+0..7 (D row = vgpr), lane 16 writes rows
        // row0+8..15 (D row = vgpr+8). e_k reconstructed from LDS: -0.5*(-2c).
        if (col == 0) {
            const int rbase = row0 + (hi ? 8 : 0);
#pragma unroll
            for (int q = 0; q < 8; ++q) {
                int row = rbase + q;
                if (row < N) {
                    int k = mini[q];
                    const float* cp = ldsB + (k >> 4) * 64 + 2 * (k & 15);
                    v2f e;
                    e.x = -0.5f * cp[0];
                    e.y = -0.5f * cp[1];
                    *(v2f*)(out_e + 2 * row) = e;   // global_store_b64
                    out_ids[row] = (float)k;
                }
            }
        }
    }
}

extern "C" void kernel_launch(void* const* d_in, const int* in_sizes, int n_in,
                              void* d_out, int out_size, void* d_ws, size_t ws_size,
                              hipStream_t stream) {
    const float* x  = (const float*)d_in[0];   // [N,2] f32
    const float* cb = (const float*)d_in[1];   // [K,2] f32

    const int N = in_sizes[0] / 2;
    const int K = in_sizes[1] / 2;

    float* out_e   = (float*)d_out;            // first output: e_k, N*2 floats
    float* out_ids = out_e + (size_t)2 * N;    // second output: ids, N values

    const int numTiles      = (N + 15) / 16;
    const int tilesPerBlock = WAVES_PER_BLOCK * TILES_PER_WAVE;
    const int blocks        = (numTiles + tilesPerBlock - 1) / tilesPerBlock;

    vq_wmma_kernel<<<blocks, 256, 0, stream>>>(x, cb, out_e, out_ids, N, K);
}